// SideWindowBoxFilter_15315853377628
// MI455X (gfx1250) — compile-verified
//
#include <hip/hip_runtime.h>
#include <hip/hip_bf16.h>

// Side-window box filter, RADIUS=1, ITERATION=5, on 1024x1024x3 f32.
// Padded working image: 1026x1026x3 f32 (12.6 MB) -> the ping-pong pair is
// L2-resident on MI455X (192 MB L2); the problem is launch/L2-latency bound,
// not FLOP bound, so the right tool is an LDS-tiled f32 stencil with CDNA5
// async global->LDS halo staging (ASYNCcnt path), wave32-friendly 256-thread
// blocks, 32x32-pixel tiles (13% halo overhead, 1089 blocks/iter).

#define IMG_H 1024
#define IMG_W 1024
#define HP    1026
#define WP    1026
#define NCH   3
#define TILE  32            // pixels per tile edge
#define BDIM  16            // 16x16 = 256 threads, 2x2 pixels per thread
#define HALO  (TILE + 2)    // 34
#define NELEM (HALO * HALO) // 1156 halo pixels
#define NPASS ((NELEM + BDIM * BDIM - 1) / (BDIM * BDIM))  // 5

// ---- CDNA5 async global->LDS path (guarded; falls back to plain loads) ----
#if defined(__gfx1250__) && __has_builtin(__builtin_amdgcn_global_load_async_to_lds_b32)
#define USE_ASYNC_LDS 1
#else
#define USE_ASYNC_LDS 0
#endif

#if USE_ASYNC_LDS
typedef __attribute__((address_space(1))) int gas1_int;  // global
typedef __attribute__((address_space(3))) int las3_int;  // LDS

__device__ __forceinline__ void async_copy_b32(const float* g, float* l) {
  __builtin_amdgcn_global_load_async_to_lds_b32(
      (gas1_int*)(g), (las3_int*)(l), /*offset=*/0, /*cpol=*/0);
}
__device__ __forceinline__ void wait_async0() {
#if __has_builtin(__builtin_amdgcn_s_wait_asynccnt)
  __builtin_amdgcn_s_wait_asynccnt(0);
#else
  asm volatile("s_wait_asynccnt 0" ::: "memory");
#endif
}
#endif

__device__ __forceinline__ int iclamp(int v, int lo, int hi) {
  return v < lo ? lo : (v > hi ? hi : v);
}

// Per-pixel side-window update from the LDS tile. Pixel at halo coords
// (py+1, px+1); returns U + d[argmin |d|] for all 3 channels.
__device__ __forceinline__ void swf_pixel(const float* __restrict__ tile,
                                          int py, int px, float res[NCH]) {
  const float half = 0.5f, third = 1.0f / 3.0f;
#pragma unroll
  for (int c = 0; c < NCH; ++c) {
    float v[3][3];
#pragma unroll
    for (int dy = 0; dy < 3; ++dy)
#pragma unroll
      for (int dx = 0; dx < 3; ++dx)
        v[dy][dx] = tile[((py + dy) * HALO + (px + dx)) * NCH + c];

    // Vertical partial sums per column: L=[.5,.5,0], R=[0,.5,.5], K=mean3.
    float sL[3], sR[3], sK[3];
#pragma unroll
    for (int dx = 0; dx < 3; ++dx) {
      const float a = v[0][dx], b = v[1][dx], e = v[2][dx];
      sL[dx] = half * (a + b);
      sR[dx] = half * (b + e);
      sK[dx] = (a + b + e) * third;
    }
    const float ctr = v[1][1];

    // 8 side windows: kernel1=[L,L,R,R,L,R,K,K], kernel2=[L,R,L,R,K,K,L,R].
    float d[8];
    d[0] = half * (sL[0] + sL[1]) - ctr;
    d[1] = half * (sL[1] + sL[2]) - ctr;
    d[2] = half * (sR[0] + sR[1]) - ctr;
    d[3] = half * (sR[1] + sR[2]) - ctr;
    d[4] = (sL[0] + sL[1] + sL[2]) * third - ctr;
    d[5] = (sR[0] + sR[1] + sR[2]) * third - ctr;
    d[6] = half * (sK[0] + sK[1]) - ctr;
    d[7] = half * (sK[1] + sK[2]) - ctr;

    // argmin |d| (first occurrence, like jnp.argmin) -> strict '<'.
    float best = d[0], ba = fabsf(d[0]);
#pragma unroll
    for (int j = 1; j < 8; ++j) {
      const float aj = fabsf(d[j]);
      if (aj < ba) { ba = aj; best = d[j]; }
    }
    res[c] = ctr + best;
  }
}

// ---- one side-window iteration ----
// final==0: dst is the padded [HP][WP][3] ping-pong buffer.
// final!=0: dst is d_out [1024][1024][3], cropped and scaled by 255.
__global__ __launch_bounds__(256) void swf_iter(const float* __restrict__ src,
                                                float* __restrict__ dst,
                                                int final) {
  __shared__ float tile[NELEM * NCH];   // 34*34*3 f32 = 13.9 KB

  const int tx = threadIdx.x, ty = threadIdx.y;
  const int bx = blockIdx.x * TILE, by = blockIdx.y * TILE;
  const int tid = ty * BDIM + tx;

  // Stage the 34x34 halo (x3 channels) into LDS: branch-free async b32
  // copies from CLAMPED coordinates (always a valid address), fully
  // unrolled (passes 0..3 unconditional, pass 4 = tid<132 tail).
#pragma unroll
  for (int p = 0; p < NPASS; ++p) {
    const int i = tid + p * (BDIM * BDIM);
    if (p + 1 < NPASS || i < NELEM) {
      const int ly = i / HALO, lx = i - ly * HALO;
      const int gy = iclamp(by + ly - 1, 0, HP - 1);
      const int gx = iclamp(bx + lx - 1, 0, WP - 1);
      const float* g = src + ((size_t)gy * WP + gx) * NCH;
      float* s = tile + i * NCH;
#if USE_ASYNC_LDS
      async_copy_b32(g + 0, s + 0);
      async_copy_b32(g + 1, s + 1);
      async_copy_b32(g + 2, s + 2);
#else
      s[0] = g[0]; s[1] = g[1]; s[2] = g[2];
#endif
    }
  }
#if USE_ASYNC_LDS
  wait_async0();
#endif

  // Fix-up: conv uses ZERO padding outside the padded array, so overwrite
  // out-of-range halo entries with 0 (only border tiles have any; interior
  // tiles predicate all-false). Issued after the async drain, before the
  // barrier -> DScnt ordering via __syncthreads.
#pragma unroll
  for (int p = 0; p < NPASS; ++p) {
    const int i = tid + p * (BDIM * BDIM);
    if (p + 1 < NPASS || i < NELEM) {
      const int ly = i / HALO, lx = i - ly * HALO;
      const int gy = by + ly - 1, gx = bx + lx - 1;
      if (gy < 0 || gy >= HP || gx < 0 || gx >= WP) {
        float* s = tile + i * NCH;
        s[0] = 0.0f; s[1] = 0.0f; s[2] = 0.0f;
      }
    }
  }
  __syncthreads();

  // 2x2 pixels per thread in quadrant layout (+0/+16) so lane-adjacent
  // pixels stay memory-adjacent -> coalesced b96 stores.
#pragma unroll
  for (int sy = 0; sy < 2; ++sy) {
#pragma unroll
    for (int sx = 0; sx < 2; ++sx) {
      const int px = tx + sx * BDIM;    // 0..31 in tile
      const int py = ty + sy * BDIM;
      const int x = bx + px, y = by + py;
      if (x >= WP || y >= HP) continue;

      float res[NCH];
      swf_pixel(tile, py, px, res);

      if (!final) {
        float* d = dst + ((size_t)y * WP + x) * NCH;
        d[0] = res[0]; d[1] = res[1]; d[2] = res[2];
      } else if (y >= 1 && y <= IMG_H && x >= 1 && x <= IMG_W) {
        float* d = dst + (((size_t)(y - 1)) * IMG_W + (x - 1)) * NCH;
        d[0] = res[0] * 255.0f; d[1] = res[1] * 255.0f; d[2] = res[2] * 255.0f;
      }
    }
  }
}

// ---- init: U0 = edge_pad(img / 255), layout [HP][WP][3] ----
__global__ __launch_bounds__(256) void swf_init(const float* __restrict__ img,
                                                float* __restrict__ U) {
  int x = blockIdx.x * BDIM + threadIdx.x;
  int y = blockIdx.y * BDIM + threadIdx.y;
  if (x >= WP || y >= HP) return;
  int sy = iclamp(y - 1, 0, IMG_H - 1);
  int sx = iclamp(x - 1, 0, IMG_W - 1);
  const float inv255 = 1.0f / 255.0f;
  const float* s = img + ((size_t)sy * IMG_W + sx) * NCH;
  float* d = U + ((size_t)y * WP + x) * NCH;
#pragma unroll
  for (int c = 0; c < NCH; ++c) d[c] = s[c] * inv255;
}

extern "C" void kernel_launch(void* const* d_in, const int* in_sizes, int n_in,
                              void* d_out, int out_size, void* d_ws, size_t ws_size,
                              hipStream_t stream) {
  const float* img = (const float*)d_in[0];   // [1024,1024,3] f32 (w1/w2 baked in)
  float* out = (float*)d_out;                 // [1024,1024,3] f32

  float* A = (float*)d_ws;                    // padded U, buffer A
  float* B = A + (size_t)HP * WP * NCH;       // padded U, buffer B

  dim3 iblk(BDIM, BDIM);
  dim3 igrd((WP + BDIM - 1) / BDIM, (HP + BDIM - 1) / BDIM);
  swf_init<<<igrd, iblk, 0, stream>>>(img, A);

  dim3 blk(BDIM, BDIM);
  dim3 grd((WP + TILE - 1) / TILE, (HP + TILE - 1) / TILE);
  swf_iter<<<grd, blk, 0, stream>>>(A, B, 0);    // iter 1
  swf_iter<<<grd, blk, 0, stream>>>(B, A, 0);    // iter 2
  swf_iter<<<grd, blk, 0, stream>>>(A, B, 0);    // iter 3
  swf_iter<<<grd, blk, 0, stream>>>(B, A, 0);    // iter 4
  swf_iter<<<grd, blk, 0, stream>>>(A, out, 1);  // iter 5 (fused crop * 255)
}